// GPU_layer_34600256536921
// MI455X (gfx1250) — compile-verified
//
#include <hip/hip_runtime.h>
#include <stdint.h>

#ifndef __has_builtin
#define __has_builtin(x) 0
#endif

#define LDS_AS __attribute__((address_space(3)))
#define GLB_AS __attribute__((address_space(1)))

#define PN      8192      // row length (P)
#define TPB     256       // threads per block (8 wave32)
#define EPT     32        // elements per thread
#define RADIX   16        // 4-bit digits
#define PASSES  8         // 8 * 4 = 32 key bits

// LDS map (bytes):                                       size
//   bufA  u64[8192]        @ 0        (aliased later by cLds f32[8192])   65536
//   bufB  u64[8192]        @ 65536    (aliased by xStage/rStage, rank16)  65536
//   hist  u32[16*256]      @ 131072                                       16384
//   chunkSum u32[256]      @ 147456                                        1024
#define SMEM_BYTES 148480

// ---- CDNA5 async global->LDS copy (ASYNCcnt-tracked) -----------------------
__device__ __forceinline__ void async_copy_b32(const float* g, float* l) {
#if __has_builtin(__builtin_amdgcn_global_load_async_to_lds_b32)
  __builtin_amdgcn_global_load_async_to_lds_b32(
      (GLB_AS int*)(void*)const_cast<float*>(g),
      (LDS_AS int*)l, /*offset=*/0, /*cpol=*/0);
#else
  unsigned lofs = (unsigned)(uintptr_t)(LDS_AS void*)l;   // 32-bit LDS byte address
  asm volatile("global_load_async_to_lds_b32 %0, %1, off"
               :: "v"(lofs), "v"(g) : "memory");
#endif
}

__device__ __forceinline__ void wait_async_zero() {
#if __has_builtin(__builtin_amdgcn_s_wait_asynccnt)
  __builtin_amdgcn_s_wait_asynccnt(0);
#else
  asm volatile("s_wait_asynccnt 0" ::: "memory");
#endif
}

// XOR bank swizzle for the u64 buffers: bijective involution on [0, 8192).
// A thread's serial walk over its contiguous chunk j = 32*t + i then hits
// bank-pair (i ^ lane) -> conflict-free across the 32 lanes of a wave.
__device__ __forceinline__ int phi(int j) { return j ^ ((j >> 5) & 31); }

__global__ void __launch_bounds__(TPB, 4)
qp_sort_rank_kernel(const float* __restrict__ x, const float* __restrict__ rho,
                    const float* __restrict__ c, float* __restrict__ out)
{
  extern __shared__ char smem[];
  unsigned long long* bufA   = (unsigned long long*)(smem);
  unsigned long long* bufB   = (unsigned long long*)(smem + 65536);
  float*              xStage = (float*)(smem + 65536);           // alias bufB[0:32K)
  float*              rStage = (float*)(smem + 98304);           // alias bufB[32K:64K)
  unsigned short*     rank16 = (unsigned short*)(smem + 65536);  // alias bufB (16K)
  float*              cLds   = (float*)(smem);                   // alias bufA (32K)
  unsigned*           hist   = (unsigned*)(smem + 131072);
  unsigned*           chunkSum = (unsigned*)(smem + 147456);

  const int  t    = (int)threadIdx.x;
  const long row  = (long)blockIdx.x;
  const float* xrow = x   + row * PN;
  const float* rrow = rho + row * PN;
  float*       orow = out + row * PN;

  // ---- stage x, rho rows into LDS via async-to-LDS loads (coalesced) ------
  for (int i = 0; i < EPT; ++i) {
    int j = t + i * TPB;
    async_copy_b32(xrow + j, xStage + j);
    async_copy_b32(rrow + j, rStage + j);
  }
  wait_async_zero();
  __syncthreads();

  // ---- pack (sortable_key << 32) | index into bufA ------------------------
  for (int i = 0; i < EPT; ++i) {
    int j = t + i * TPB;
    unsigned u = __float_as_uint(xStage[j] * rStage[j]);
    u ^= ((u >> 31) ? 0xFFFFFFFFu : 0x80000000u);   // float -> ascending uint
    bufA[phi(j)] = ((unsigned long long)u << 32) | (unsigned)j;
  }
  __syncthreads();

  // ---- stable LSD radix sort: 8 passes of 4 bits over the key -------------
  // Per-thread histogram state lives in VGPRs (16 bins x 8-bit fields in 4
  // u32 regs; counts <= 32). Per-element running counts are captured in the
  // count loop into statically-indexed packed bytes (run_pack). Elements are
  // NOT kept in registers across the scan (reloaded from LDS in the scatter)
  // so total allocation stays well under 256 VGPRs: no scratch, 4 waves/SIMD.
  const int base = t * EPT;  // contiguous logical chunk => stability
  for (int pass = 0; pass < PASSES; ++pass) {
    unsigned long long* src = (pass & 1) ? bufB : bufA;
    unsigned long long* dst = (pass & 1) ? bufA : bufB;
    const int shift = pass * 4;

    // count digits + record per-element running counts (packed bytes)
    unsigned run_pack[EPT / 4] = {0, 0, 0, 0, 0, 0, 0, 0};
    unsigned c0 = 0, c1 = 0, c2 = 0, c3 = 0;
#pragma unroll
    for (int i = 0; i < EPT; ++i) {
      unsigned hi  = (unsigned)(src[phi(base + i)] >> 32);
      unsigned d   = (hi >> shift) & 15u;
      unsigned sh  = (d & 3u) * 8u;
      unsigned sel = d >> 2;
      unsigned packed = (sel == 0u) ? c0 : ((sel == 1u) ? c1 : ((sel == 2u) ? c2 : c3));
      unsigned run = (packed >> sh) & 0xFFu;           // pre-increment count
      run_pack[i >> 2] |= run << ((i & 3) * 8);        // static index -> regs
      unsigned inc = 1u << sh;
      c0 += (sel == 0u) ? inc : 0u;
      c1 += (sel == 1u) ? inc : 0u;
      c2 += (sel == 2u) ? inc : 0u;
      c3 += (sel == 3u) ? inc : 0u;
    }
    // publish bins: hist[d*256+t] (bank = t -> conflict-free); no zeroing pass
#pragma unroll
    for (int d = 0; d < RADIX; ++d) {
      unsigned reg = (d >> 2) == 0 ? c0 : ((d >> 2) == 1 ? c1 : ((d >> 2) == 2 ? c2 : c3));
      hist[d * TPB + t] = (reg >> ((d & 3) * 8)) & 0xFFu;
    }
    __syncthreads();

    // ---- exclusive scan of 4096 counters, flat (digit-major, thread-minor).
    // Level 1: each thread scans its 16 counters as 4 sequential uint4 groups
    {
      const int f0 = t * RADIX;
      unsigned run = 0;
#pragma unroll
      for (int g = 0; g < 4; ++g) {
        uint4 a = *(uint4*)(hist + f0 + 4 * g);
        unsigned t0 = a.x, t1 = a.y, t2 = a.z, t3 = a.w;
        a.x = run; run += t0;
        a.y = run; run += t1;
        a.z = run; run += t2;
        a.w = run; run += t3;
        *(uint4*)(hist + f0 + 4 * g) = a;
      }
      chunkSum[t] = run;
    }
    __syncthreads();
    // Level 2: wave 0 scans all 256 chunk sums (8/lane + wave32 shuffle scan)
    if (t < 32) {
      uint4 b0 = *(uint4*)(chunkSum + t * 8);
      uint4 b1 = *(uint4*)(chunkSum + t * 8 + 4);
      unsigned run = 0, tv;
      tv = b0.x; b0.x = run; run += tv;
      tv = b0.y; b0.y = run; run += tv;
      tv = b0.z; b0.z = run; run += tv;
      tv = b0.w; b0.w = run; run += tv;
      tv = b1.x; b1.x = run; run += tv;
      tv = b1.y; b1.y = run; run += tv;
      tv = b1.z; b1.z = run; run += tv;
      tv = b1.w; b1.w = run; run += tv;
      unsigned incl = run;
#pragma unroll
      for (int ofs = 1; ofs < 32; ofs <<= 1) {
        unsigned n = __shfl_up(incl, ofs, 32);
        if (t >= ofs) incl += n;
      }
      unsigned basev = incl - run;   // exclusive prefix of this lane's group
      b0.x += basev; b0.y += basev; b0.z += basev; b0.w += basev;
      b1.x += basev; b1.y += basev; b1.z += basev; b1.w += basev;
      *(uint4*)(chunkSum + t * 8)     = b0;
      *(uint4*)(chunkSum + t * 8 + 4) = b1;
    }
    __syncthreads();
    // Add-back: hist[f] += chunkSum[f>>4] (sequential uint4 groups)
    {
      const int f0 = t * RADIX;
      const unsigned add = chunkSum[t];
#pragma unroll
      for (int g = 0; g < 4; ++g) {
        uint4 a = *(uint4*)(hist + f0 + 4 * g);
        a.x += add; a.y += add; a.z += add; a.w += add;
        *(uint4*)(hist + f0 + 4 * g) = a;
      }
    }
    __syncthreads();

    // stable scatter: reload element (pipelined ds_load_b64), base (ds_load)
    // + recorded run count (static-index packed bytes)
#pragma unroll
    for (int i = 0; i < EPT; ++i) {
      unsigned long long e = src[phi(base + i)];
      unsigned d   = ((unsigned)(e >> 32) >> shift) & 15u;
      unsigned run = (run_pack[i >> 2] >> ((i & 3) * 8)) & 0xFFu;
      unsigned pos = hist[d * TPB + t] + run;
      dst[phi((int)pos)] = e;
    }
    __syncthreads();
  }

  // ---- sorted pairs are in bufA; build rank table (aliases dead bufB) -----
  for (int i = 0; i < EPT; ++i) {
    int p = t + i * TPB;
    unsigned idx = (unsigned)bufA[phi(p)];     // low 32 bits = original index
    rank16[idx] = (unsigned short)p;
  }
  __syncthreads();

  // ---- stage c into LDS (aliases dead bufA) via async loads ---------------
  for (int i = 0; i < EPT; ++i) {
    int j = t + i * TPB;
    async_copy_b32(c + j, cLds + j);
  }
  wait_async_zero();
  __syncthreads();

  // ---- out[j] = max(x[j], -c[rank(j)]), coalesced float4 I/O --------------
  for (int i = 0; i < EPT / 4; ++i) {
    int j0 = (t + i * TPB) * 4;
    float4 xv = *(const float4*)(xrow + j0);
    float4 r;
    r.x = fmaxf(xv.x, -cLds[rank16[j0 + 0]]);
    r.y = fmaxf(xv.y, -cLds[rank16[j0 + 1]]);
    r.z = fmaxf(xv.z, -cLds[rank16[j0 + 2]]);
    r.w = fmaxf(xv.w, -cLds[rank16[j0 + 3]]);
    *(float4*)(orow + j0) = r;
  }
}

extern "C" void kernel_launch(void* const* d_in, const int* in_sizes, int n_in,
                              void* d_out, int out_size, void* d_ws, size_t ws_size,
                              hipStream_t stream) {
  (void)n_in; (void)d_ws; (void)ws_size; (void)out_size;
  const float* x   = (const float*)d_in[0];
  const float* rho = (const float*)d_in[1];
  const float* c   = (const float*)d_in[2];
  float* out = (float*)d_out;
  const int B = in_sizes[0] / PN;   // 4096 rows

  // gfx1250 allows up to 320 KB LDS per workgroup; opt in for 148.5 KB.
  (void)hipFuncSetAttribute((const void*)qp_sort_rank_kernel,
                            hipFuncAttributeMaxDynamicSharedMemorySize,
                            (int)SMEM_BYTES);

  qp_sort_rank_kernel<<<dim3(B), dim3(TPB), SMEM_BYTES, stream>>>(x, rho, c, out);
}